// Soft_NECTAR_Binning_48782238548323
// MI455X (gfx1250) — compile-verified
//
#include <hip/hip_runtime.h>
#include <stdint.h>

#define NCLS        8
#define NLOC        9
#define NPROB       15
#define IMG_H       512
#define IMG_W       512
#define HW          (IMG_H * IMG_W)
#define TW          32
#define TH          8
#define HALO_W      (TW + 2)      // 34
#define HALO_H      (TH + 2)      // 10
#define ROWSTRIDE   (TW + 4)      // 36 floats: conflict-free stencil reads
#define VF_ELEMS    (NCLS * NLOC * NPROB)   // 1080 floats = 270 x 16B

__global__ __launch_bounds__(TW * TH)
void soft_nectar_binning_kernel(const float* __restrict__ logits,
                                const float* __restrict__ val_freqs,
                                float* __restrict__ out)
{
    __shared__ __align__(16) float s_vf[VF_ELEMS];
    __shared__ float s_probs[NCLS][HALO_H][ROWSTRIDE];

    const int tx  = threadIdx.x;          // 0..31
    const int ty  = threadIdx.y;          // 0..7
    const int tid = ty * TW + tx;         // 0..255
    const int w0  = blockIdx.x * TW;
    const int h0  = blockIdx.y * TH;
    const int b   = blockIdx.z;

    // ---------------- Phase 0: async DMA the calibration table into LDS ----
    // 270 x b128 transfers; overlaps with phase-1 softmax below.
    {
        const uint32_t lds0 = (uint32_t)(uintptr_t)(&s_vf[0]);  // low 32b of
                                                                // generic LDS
                                                                // ptr = offset
        uint32_t       l    = lds0 + (uint32_t)tid * 16u;
        const char*    gp   = (const char*)val_freqs + (size_t)tid * 16u;
        asm volatile("global_load_async_to_lds_b128 %0, %1, off"
                     :: "v"(l), "v"(gp) : "memory");
        if (tid < (VF_ELEMS / 4) - 256) {   // tid < 14
            uint32_t    l2  = l  + 256u * 16u;
            const char* gp2 = gp + 256u * 16u;
            asm volatile("global_load_async_to_lds_b128 %0, %1, off"
                         :: "v"(l2), "v"(gp2) : "memory");
        }
    }

    // ---------------- Phase 1: softmax probs for the 34x10 halo tile -------
    const float* lbase = logits + (size_t)b * (NCLS * HW);
    for (int p = tid; p < HALO_W * HALO_H; p += TW * TH) {
        const int py = p / HALO_W;
        const int px = p - py * HALO_W;
        const int hy = h0 + py - 1;
        const int wx = w0 + px - 1;
        if (hy >= 0 && hy < IMG_H && wx >= 0 && wx < IMG_W) {
            const float* lp = lbase + (size_t)hy * IMG_W + wx;
            float x[NCLS];
            #pragma unroll
            for (int c = 0; c < NCLS; ++c) x[c] = lp[(size_t)c * HW];
            float m = x[0];
            #pragma unroll
            for (int c = 1; c < NCLS; ++c) m = fmaxf(m, x[c]);
            float s = 0.0f;
            #pragma unroll
            for (int c = 0; c < NCLS; ++c) { x[c] = __expf(x[c] - m); s += x[c]; }
            const float r = 1.0f / s;
            #pragma unroll
            for (int c = 0; c < NCLS; ++c) s_probs[c][py][px] = x[c] * r;
        } else {
            // SAME padding in reduce_window pads with the init value 0.0
            #pragma unroll
            for (int c = 0; c < NCLS; ++c) s_probs[c][py][px] = 0.0f;
        }
    }

    // Table DMA must have landed; then make all probs visible.
    asm volatile("s_wait_asynccnt 0" ::: "memory");
    __syncthreads();

    // ---------------- Phase 2: stencil + binning + gather + normalize ------
    float cal[NCLS];
    float csum = 0.0f;
    #pragma unroll
    for (int c = 0; c < NCLS; ++c) {
        float n = 0.0f, ctr = 0.0f;
        #pragma unroll
        for (int dy = 0; dy < 3; ++dy) {
            #pragma unroll
            for (int dx = 0; dx < 3; ++dx) {
                const float v = s_probs[c][ty + dy][tx + dx];
                n += v;
                if (dy == 1 && dx == 1) ctr = v;
            }
        }
        // reference: neigh_mean = sum / 9.0 ; loc = clip(floor(mean * 9))
        const float nm  = n / 9.0f;
        int loc = (int)floorf(nm * 9.0f);
        loc = min(max(loc, 0), NLOC - 1);
        int pb = (int)floorf(ctr * 15.0f);
        pb = min(max(pb, 0), NPROB - 1);
        const float v = s_vf[(c * NLOC + loc) * NPROB + pb];
        cal[c] = v;
        csum  += v;
    }
    if (csum == 0.0f) csum = 1.0f;   // where(s==0, 1, s)
    const float r = 1.0f / csum;

    float* ob = out + (size_t)b * (NCLS * HW) + (size_t)(h0 + ty) * IMG_W + (w0 + tx);
    #pragma unroll
    for (int c = 0; c < NCLS; ++c) ob[(size_t)c * HW] = cal[c] * r;
}

extern "C" void kernel_launch(void* const* d_in, const int* in_sizes, int n_in,
                              void* d_out, int out_size, void* d_ws, size_t ws_size,
                              hipStream_t stream)
{
    (void)n_in; (void)d_ws; (void)ws_size; (void)out_size;
    const float* logits    = (const float*)d_in[0];   // [B, 8, 512, 512] f32
    const float* val_freqs = (const float*)d_in[1];   // [8, 9, 15] f32
    float*       out       = (float*)d_out;           // [B, 8, 512, 512] f32

    const int B = in_sizes[0] / (NCLS * HW);          // 16

    dim3 grid(IMG_W / TW, IMG_H / TH, B);             // 16 x 64 x B
    dim3 block(TW, TH, 1);                            // 256 threads = 8 waves
    soft_nectar_binning_kernel<<<grid, block, 0, stream>>>(logits, val_freqs, out);
}